// RoI3DPool_15006615733621
// MI455X (gfx1250) — compile-verified
//
#include <hip/hip_runtime.h>

// Problem constants (fixed by the reference's setup_inputs).
#define PD 7
#define PH 7
#define PW 7
#define PP (PD * PH * PW)      // 343 pooled cells per (roi, channel)
#define BATCH 2
#define CH 256
#define DDIM 32
#define HDIM 64
#define WDIM 64
#define HW   (HDIM * WDIM)     // 4096
#define DHW  (DDIM * HW)       // 131072 floats: channel stride
#define CDHW (CH * DHW)        // 33554432 floats: batch stride

// gfx1250 async global->LDS path (ASYNCcnt-tracked, bypasses VGPRs).
#if __has_builtin(__builtin_amdgcn_global_load_async_to_lds_b32) && \
    __has_builtin(__builtin_amdgcn_s_wait_asynccnt)
#define USE_ASYNC_LDS 1
#else
#define USE_ASYNC_LDS 0
#endif

typedef __attribute__((address_space(1))) int* gas_i32_ptr;  // global AS
typedef __attribute__((address_space(3))) int* las_i32_ptr;  // LDS AS

// One block per ROI. 352 threads = 11 wave32 waves; threads 0..342 each own
// one (z,y,x) pooled cell and loop over the 256 channels.
__global__ __launch_bounds__(352) void roi3d_gather_kernel(
    const float* __restrict__ features,   // (B, C, D, H, W)
    const float* __restrict__ rois,       // (N, 7) = [b, x1, y1, z1, x2, y2, z2]
    float* __restrict__ out)              // (N, C, 7, 7, 7)
{
    __shared__ float s_roi[7];
    __shared__ int s_ix[PW];
    __shared__ int s_iy[PH];
    __shared__ int s_iz[PD];
    __shared__ int s_boff;

    const int n = blockIdx.x;
    const int t = threadIdx.x;
    const float* roi = rois + (size_t)n * 7;

    // ---- Phase 0: stage the 7 ROI floats into LDS ----------------------
#if USE_ASYNC_LDS
    if (t < 7) {
        // global_load_async_to_lds_b32: per-lane global src -> per-lane LDS dst
        float* gp = const_cast<float*>(roi + t);     // builtin takes non-const
        int*   lp = (int*)&s_roi[t];
        __builtin_amdgcn_global_load_async_to_lds_b32(
            (gas_i32_ptr)gp, (las_i32_ptr)lp,
            /*offset=*/0, /*cpol=*/0);
    }
    __builtin_amdgcn_s_wait_asynccnt(0);
#else
    if (t < 7) {
        s_roi[t] = roi[t];
    }
#endif
    __syncthreads();

    // ---- Phase 1: per-ROI index tables (21 threads, one index each) ----
    if (t == 0) {
        // batch index stored as float; truncation matches astype(int32)
        s_boff = (int)s_roi[0] * CDHW;
    }
    if (t < 3 * PD) {
        const int dim = t / 7;          // 0 -> x, 1 -> y, 2 -> z
        const int p   = t - dim * 7;
        float lo, hi;
        int lim;
        if (dim == 0)      { lo = s_roi[1]; hi = s_roi[4]; lim = WDIM - 1; }
        else if (dim == 1) { lo = s_roi[2]; hi = s_roi[5]; lim = HDIM - 1; }
        else               { lo = s_roi[3]; hi = s_roi[6]; lim = DDIM - 1; }
        // SPATIAL_SCALE == 1.0
        const float flo = fminf(fmaxf(lo, 0.0f), (float)lim);
        const float fhi = fminf(fmaxf(hi, 0.0f), (float)lim);
        const float tt  = (float)p * (1.0f / (float)(PD - 1));
        int idx = (int)floorf(flo + (fhi - flo) * tt);
        idx = min(max(idx, 0), lim);
        if (dim == 0)      s_ix[p] = idx;
        else if (dim == 1) s_iy[p] = idx;
        else               s_iz[p] = idx;
    }
    __syncthreads();

    if (t >= PP) return;

    // ---- Phase 2: gather loop ------------------------------------------
    // Decompose pooled-cell id: t = z*49 + y*7 + x  (x fastest -> coalesced out)
    const int z = t / (PH * PW);
    const int r = t - z * (PH * PW);
    const int y = r / PW;
    const int x = r - y * PW;

    const int foff = s_boff + s_iz[z] * HW + s_iy[y] * WDIM + s_ix[x];
    const float* fp = features + foff;                  // advances by DHW per channel
    float*       op = out + (size_t)n * (CH * PP) + t;  // advances by PP per channel

    // Channel loop: 1 gather load + 1 coalesced store per channel.
    // Prefetch 8 channels ahead (speculative gfx1250 global_prefetch_b8) to
    // keep deep load queues against HBM latency.
    #pragma unroll 4
    for (int c = 0; c < CH; ++c) {
        if (c + 8 < CH) {
            __builtin_prefetch(fp + (size_t)(c + 8) * DHW, 0, 1);
        }
        op[(size_t)c * PP] = fp[(size_t)c * DHW];
    }
}

extern "C" void kernel_launch(void* const* d_in, const int* in_sizes, int n_in,
                              void* d_out, int out_size, void* d_ws, size_t ws_size,
                              hipStream_t stream) {
    const float* features = (const float*)d_in[0];
    const float* rois     = (const float*)d_in[1];
    float*       out      = (float*)d_out;

    const int N = in_sizes[1] / 7;   // 512 ROIs

    roi3d_gather_kernel<<<dim3(N), dim3(352), 0, stream>>>(features, rois, out);
}